// DecoderLSTM_29764123361289
// MI455X (gfx1250) — compile-verified
//
#include <hip/hip_runtime.h>
#include <math.h>

#define H 1024
#define V 50257
#define NTILES 3142          // ceil(V/16); only tile 3141 is partial (1 row)
#define NPART 128            // softmax partial blocks

typedef __attribute__((ext_vector_type(2))) float v2f;
typedef __attribute__((ext_vector_type(8))) float v8f;

// ---------------------------------------------------------------------------
// Kernel 1: gates[r] = W_ih[r]@x + b_ih[r] + W_hh[r]@h0 + b_hh[r], r in [0,4096)
// One wave32 per row; float4 streaming loads; shuffle reduction.
// ---------------------------------------------------------------------------
__global__ void lstm_gates_kernel(const float* __restrict__ emb,
                                  const float* __restrict__ W_ih,
                                  const float* __restrict__ W_hh,
                                  const float* __restrict__ b_ih,
                                  const float* __restrict__ b_hh,
                                  const float* __restrict__ h0,
                                  const int*   __restrict__ input_idx,
                                  float* __restrict__ gates) {
    const int wave = threadIdx.x >> 5;
    const int lane = threadIdx.x & 31;
    const int row  = blockIdx.x * 8 + wave;              // 512 blocks * 8 waves = 4096

    const float* __restrict__ x  = emb + (size_t)input_idx[0] * H;
    const float* __restrict__ wi = W_ih + (size_t)row * H;
    const float* __restrict__ wh = W_hh + (size_t)row * H;

    float sum = 0.0f;
    #pragma unroll
    for (int kk = 0; kk < 8; ++kk) {
        const int col = kk * 128 + lane * 4;             // 32 lanes * 4 = 128 per step
        float4 wv = *(const float4*)(wi + col);
        float4 xv = *(const float4*)(x  + col);
        sum += wv.x * xv.x + wv.y * xv.y + wv.z * xv.z + wv.w * xv.w;
        float4 hv  = *(const float4*)(wh + col);
        float4 h0v = *(const float4*)(h0 + col);
        sum += hv.x * h0v.x + hv.y * h0v.y + hv.z * h0v.z + hv.w * h0v.w;
    }
    #pragma unroll
    for (int off = 16; off > 0; off >>= 1)
        sum += __shfl_xor(sum, off, 32);
    if (lane == 0)
        gates[row] = sum + b_ih[row] + b_hh[row];
}

// ---------------------------------------------------------------------------
// Kernel 2: LSTM cell activations. Gate order: i, f, g, o.
// ---------------------------------------------------------------------------
__global__ void lstm_cell_kernel(const float* __restrict__ gates,
                                 const float* __restrict__ c0,
                                 float* __restrict__ h_out,   // d_out + V
                                 float* __restrict__ c_out,   // d_out + V + H
                                 float* __restrict__ h_ws) {  // workspace copy
    const int j = blockIdx.x * blockDim.x + threadIdx.x;
    if (j >= H) return;
    const float gi = gates[j];
    const float gf = gates[j + H];
    const float gg = gates[j + 2 * H];
    const float go = gates[j + 3 * H];
    const float i = 1.0f / (1.0f + expf(-gi));
    const float f = 1.0f / (1.0f + expf(-gf));
    const float g = tanhf(gg);
    const float o = 1.0f / (1.0f + expf(-go));
    const float c_new = f * c0[j] + i * g;
    const float h_new = o * tanhf(c_new);
    h_out[j] = h_new;
    c_out[j] = c_new;
    h_ws[j]  = h_new;
}

// ---------------------------------------------------------------------------
// Kernel 3: logits = W_out @ h_new + b_out  via V_WMMA_F32_16X16X4_F32.
// One wave per 16-row tile of W_out; h_new broadcast across all 16 B columns,
// so every column of D holds the same GEMV result.
// A layout (16x4 f32): lanes 0-15 hold (M=lane, K=0/1) in a.x/a.y,
//                      lanes 16-31 hold (M=lane-16, K=2/3).
// D layout: vgpr r, lanes 0-15 -> M=r; lanes 16-31 -> M=r+8.
// 4 independent accumulators break the D->C dependency chain so the WMMAs
// pipeline behind the global loads; prefetch runs ~2KB ahead on each row.
// ---------------------------------------------------------------------------
__global__ void logits_wmma_kernel(const float* __restrict__ W_out,
                                   const float* __restrict__ b_out,
                                   const float* __restrict__ h,
                                   float* __restrict__ logits) {
    __shared__ float xs[H];
    for (int i = threadIdx.x; i < H; i += blockDim.x)
        xs[i] = h[i];
    __syncthreads();

    const int wave = threadIdx.x >> 5;
    const int lane = threadIdx.x & 31;
    const int tile = blockIdx.x * 8 + wave;
    if (tile >= NTILES) return;

    int m = (tile << 4) + (lane & 15);
    if (m > V - 1) m = V - 1;                 // clamp addressing for last tile
    const int khalf = lane >> 4;              // 0 for lanes 0-15, 1 for lanes 16-31
    const float* __restrict__ wrow = W_out + (size_t)m * H + (khalf << 1);

    v8f c0 = {}, c1 = {}, c2 = {}, c3 = {};
    for (int k = 0; k < H; k += 16) {
        __builtin_prefetch(wrow + k + 512, 0, 3);   // global_prefetch_b8, ~2KB ahead
        v2f a0 = *(const v2f*)(wrow + k);
        v2f a1 = *(const v2f*)(wrow + k + 4);
        v2f a2 = *(const v2f*)(wrow + k + 8);
        v2f a3 = *(const v2f*)(wrow + k + 12);
        v2f b0, b1, b2, b3;
        b0.x = xs[k +      khalf];  b0.y = xs[k +  2 + khalf];
        b1.x = xs[k +  4 + khalf];  b1.y = xs[k +  6 + khalf];
        b2.x = xs[k +  8 + khalf];  b2.y = xs[k + 10 + khalf];
        b3.x = xs[k + 12 + khalf];  b3.y = xs[k + 14 + khalf];
        c0 = __builtin_amdgcn_wmma_f32_16x16x4_f32(false, a0, false, b0, (short)0, c0, false, false);
        c1 = __builtin_amdgcn_wmma_f32_16x16x4_f32(false, a1, false, b1, (short)0, c1, false, false);
        c2 = __builtin_amdgcn_wmma_f32_16x16x4_f32(false, a2, false, b2, (short)0, c2, false, false);
        c3 = __builtin_amdgcn_wmma_f32_16x16x4_f32(false, a3, false, b3, (short)0, c3, false, false);
    }
    v8f c = (c0 + c1) + (c2 + c3);

    // lane 0 owns rows tile*16+0..7 (c[0..7]); lane 16 owns rows +8..15.
    if ((lane & 15) == 0) {
        const int rbase = (tile << 4) + (khalf << 3);
        if (tile < NTILES - 1) {
            #pragma unroll
            for (int r = 0; r < 8; ++r)
                logits[rbase + r] = c[r] + b_out[rbase + r];
        } else {
            for (int r = 0; r < 8; ++r) {
                const int row = rbase + r;
                if (row < V) logits[row] = c[r] + b_out[row];
            }
        }
    }
}

// ---------------------------------------------------------------------------
// Kernel 4: per-block online-softmax partials (m, s) over logits.
// ---------------------------------------------------------------------------
__global__ void softmax_partial_kernel(const float* __restrict__ logits,
                                       float* __restrict__ part_m,
                                       float* __restrict__ part_s) {
    __shared__ float sm[256];
    __shared__ float ss[256];
    const int t = threadIdx.x;
    float m = -3.4e38f, s = 0.0f;
    for (int v = blockIdx.x * 256 + t; v < V; v += NPART * 256) {
        const float x = logits[v];
        if (x > m) { s = s * expf(m - x) + 1.0f; m = x; }
        else       { s += expf(x - m); }
    }
    sm[t] = m; ss[t] = s;
    __syncthreads();
    for (int off = 128; off > 0; off >>= 1) {
        if (t < off) {
            const float m1 = sm[t], s1 = ss[t];
            const float m2 = sm[t + off], s2 = ss[t + off];
            const float M = fmaxf(m1, m2);
            ss[t] = s1 * expf(m1 - M) + s2 * expf(m2 - M);
            sm[t] = M;
        }
        __syncthreads();
    }
    if (t == 0) { part_m[blockIdx.x] = sm[0]; part_s[blockIdx.x] = ss[0]; }
}

// ---------------------------------------------------------------------------
// Kernel 5: combine NPART partials -> (M, log S).
// ---------------------------------------------------------------------------
__global__ void softmax_combine_kernel(const float* __restrict__ part_m,
                                       const float* __restrict__ part_s,
                                       float* __restrict__ finalv) {
    __shared__ float sm[NPART];
    __shared__ float ss[NPART];
    const int t = threadIdx.x;
    sm[t] = part_m[t]; ss[t] = part_s[t];
    __syncthreads();
    for (int off = NPART / 2; off > 0; off >>= 1) {
        if (t < off) {
            const float m1 = sm[t], s1 = ss[t];
            const float m2 = sm[t + off], s2 = ss[t + off];
            const float M = fmaxf(m1, m2);
            ss[t] = s1 * expf(m1 - M) + s2 * expf(m2 - M);
            sm[t] = M;
        }
        __syncthreads();
    }
    if (t == 0) { finalv[0] = sm[0]; finalv[1] = logf(ss[0]); }
}

// ---------------------------------------------------------------------------
// Kernel 6: log_probs = logits - M - log S (in place over d_out[0:V)).
// ---------------------------------------------------------------------------
__global__ void logsoftmax_final_kernel(float* __restrict__ logits,
                                        const float* __restrict__ finalv) {
    const int v = blockIdx.x * blockDim.x + threadIdx.x;
    if (v < V) logits[v] = logits[v] - finalv[0] - finalv[1];
}

// ---------------------------------------------------------------------------
extern "C" void kernel_launch(void* const* d_in, const int* in_sizes, int n_in,
                              void* d_out, int out_size, void* d_ws, size_t ws_size,
                              hipStream_t stream) {
    const float* emb   = (const float*)d_in[0];
    const float* W_ih  = (const float*)d_in[1];
    const float* W_hh  = (const float*)d_in[2];
    const float* b_ih  = (const float*)d_in[3];
    const float* b_hh  = (const float*)d_in[4];
    const float* W_out = (const float*)d_in[5];
    const float* b_out = (const float*)d_in[6];
    const float* h0    = (const float*)d_in[7];
    const float* c0    = (const float*)d_in[8];
    const int*   idx   = (const int*)d_in[9];

    float* out = (float*)d_out;           // [0,V): log_probs; [V,V+H): h_new; [V+H,V+2H): c_new
    float* ws  = (float*)d_ws;
    float* gates  = ws;                   // 4096
    float* h_ws   = ws + 4096;            // 1024
    float* part_m = ws + 5120;            // 128
    float* part_s = ws + 5248;            // 128
    float* finalv = ws + 5376;            // 2

    // 1) gates (4096 rows, 8 waves/block of 256 threads)
    lstm_gates_kernel<<<512, 256, 0, stream>>>(emb, W_ih, W_hh, b_ih, b_hh, h0, idx, gates);

    // 2) cell activations -> h_new, c_new
    lstm_cell_kernel<<<(H + 255) / 256, 256, 0, stream>>>(gates, c0,
                                                          out + V, out + V + H, h_ws);

    // 3) vocab GEMV via WMMA: 3142 tiles, 8 waves/block
    logits_wmma_kernel<<<(NTILES + 7) / 8, 256, 0, stream>>>(W_out, b_out, h_ws, out);

    // 4) online-softmax partials
    softmax_partial_kernel<<<NPART, 256, 0, stream>>>(out, part_m, part_s);

    // 5) combine -> (M, logS)
    softmax_combine_kernel<<<1, NPART, 0, stream>>>(part_m, part_s, finalv);

    // 6) finalize log_probs in place
    logsoftmax_final_kernel<<<(V + 255) / 256, 256, 0, stream>>>(out, finalv);
}